// EncGAT_52793738002753
// MI455X (gfx1250) — compile-verified
//
#include <hip/hip_runtime.h>
#include <hip/hip_bf16.h>
#include <math.h>

typedef __attribute__((ext_vector_type(2))) float v2f;
typedef __attribute__((ext_vector_type(8))) float v8f;

#define NEG_SLOPE 0.2f
#define NHEADS 4
#define DHEAD 64
#define HID 256   // NHEADS * DHEAD

// ---------------------------------------------------------------------------
// helpers
// ---------------------------------------------------------------------------
__device__ __forceinline__ float lrelu(float x) {
  return x > 0.0f ? x : NEG_SLOPE * x;
}

// Native f32 atomic add (GLOBAL_ATOMIC_ADD_F32) — avoid CAS-loop expansion.
__device__ __forceinline__ void atomAddF(float* p, float v) {
#if defined(__HIP_DEVICE_COMPILE__)
  unsafeAtomicAdd(p, v);
#else
  atomicAdd(p, v);
#endif
}

// float atomic max via IEEE int-ordering trick (valid incl. -inf init):
// lowers to native global_atomic_max_i32 / global_atomic_min_u32.
__device__ __forceinline__ void atomicMaxF(float* addr, float v) {
  if (v >= 0.0f) {
    atomicMax((int*)addr, __float_as_int(v));
  } else {
    atomicMin((unsigned int*)addr, (unsigned int)__float_as_int(v));
  }
}

// ---------------------------------------------------------------------------
// GEMM: feat[N,HID] = h[N,K] @ W[K,HID], FP32 via V_WMMA_F32_16X16X4_F32.
// One wave computes a 16x64 output strip (4 accumulators) so the A fragment
// is fetched once per 64 output columns.
//
// A 16x4 f32 layout:  lanes 0-15 -> M=lane,   K={k,k+1};
//                     lanes16-31 -> M=lane-16,K={k+2,k+3}
// B 4x16 f32 layout:  vgpr0: K=k   (lanes0-15) / K=k+2 (lanes16-31), N=lane&15
//                     vgpr1: K=k+1 (lanes0-15) / K=k+3 (lanes16-31)
// C/D 16x16 f32:      vgpr r: M=r (lanes0-15) / M=r+8 (lanes16-31), N=lane&15
// ---------------------------------------------------------------------------
__global__ __launch_bounds__(256) void gemm_wmma_f32(
    const float* __restrict__ h, const float* __restrict__ W,
    float* __restrict__ feat, int N, int K) {
  const int lane = threadIdx.x & 31;
  const int wave = threadIdx.x >> 5;
  const int rowTile = blockIdx.x * 8 + wave;  // 16 rows per tile
  if (rowTile * 16 >= N) return;
  const int col0 = blockIdx.y * 64;           // 64-column strip
  const int row  = rowTile * 16 + (lane & 15);
  const int kSel = (lane >> 4) * 2;           // 0 for lanes 0-15, 2 for 16-31
  const int colL = lane & 15;

  v8f c0 = {}, c1 = {}, c2 = {}, c3 = {};
  for (int k = 0; k < K; k += 4) {
    const float* ap = h + (size_t)row * K + k + kSel;
    v2f a;
    a.x = ap[0];
    a.y = ap[1];
    const float* bp = W + (size_t)(k + kSel) * HID + col0 + colL;
    v2f b0, b1, b2, b3;
    b0.x = bp[0];  b0.y = bp[HID + 0];
    b1.x = bp[16]; b1.y = bp[HID + 16];
    b2.x = bp[32]; b2.y = bp[HID + 32];
    b3.x = bp[48]; b3.y = bp[HID + 48];
    c0 = __builtin_amdgcn_wmma_f32_16x16x4_f32(false, a, false, b0, (short)0, c0, false, false);
    c1 = __builtin_amdgcn_wmma_f32_16x16x4_f32(false, a, false, b1, (short)0, c1, false, false);
    c2 = __builtin_amdgcn_wmma_f32_16x16x4_f32(false, a, false, b2, (short)0, c2, false, false);
    c3 = __builtin_amdgcn_wmma_f32_16x16x4_f32(false, a, false, b3, (short)0, c3, false, false);
  }

  const int mBase = rowTile * 16 + (lane >> 4) * 8;
#pragma unroll
  for (int r = 0; r < 8; ++r) {
    float* op = feat + (size_t)(mBase + r) * HID + col0 + colL;
    op[0]  = c0[r];
    op[16] = c1[r];
    op[32] = c2[r];
    op[48] = c3[r];
  }
}

// ---------------------------------------------------------------------------
// el[n,h] = sum_d feat[n,h,d]*al[h,d]; er likewise. One wave per (node,head).
// ---------------------------------------------------------------------------
__global__ __launch_bounds__(256) void node_attn(
    const float* __restrict__ feat, const float* __restrict__ al,
    const float* __restrict__ ar, float* __restrict__ el,
    float* __restrict__ er, int NH) {
  const int lane = threadIdx.x & 31;
  const int nh = blockIdx.x * 8 + (threadIdx.x >> 5);
  if (nh >= NH) return;
  const int hh = nh & (NHEADS - 1);
  const float* fp  = feat + (size_t)nh * DHEAD;
  const float* alp = al + hh * DHEAD;
  const float* arp = ar + hh * DHEAD;
  float sl = fp[lane] * alp[lane] + fp[lane + 32] * alp[lane + 32];
  float sr = fp[lane] * arp[lane] + fp[lane + 32] * arp[lane + 32];
#pragma unroll
  for (int off = 16; off > 0; off >>= 1) {
    sl += __shfl_down(sl, off);
    sr += __shfl_down(sr, off);
  }
  if (lane == 0) {
    el[nh] = sl;
    er[nh] = sr;
  }
}

// ---------------------------------------------------------------------------
// zero accumulator, init segment max (-inf) and segment sum (0)
// ---------------------------------------------------------------------------
__global__ void init_bufs(float* __restrict__ accum, float* __restrict__ m,
                          float* __restrict__ s, int nOut, int nNH) {
  const int i = blockIdx.x * blockDim.x + threadIdx.x;
  if (i < nOut) accum[i] = 0.0f;
  if (i < nNH) {
    m[i] = -INFINITY;
    s[i] = 0.0f;
  }
}

// ---------------------------------------------------------------------------
// pass 1 over edges: e = leaky_relu(el[src]+er[dst]); store e; segment max
// ---------------------------------------------------------------------------
__global__ void edge_logits_max(const float* __restrict__ el,
                                const float* __restrict__ er,
                                const int* __restrict__ src,
                                const int* __restrict__ dst,
                                float* __restrict__ ebuf,
                                float* __restrict__ m, int E) {
  const int e = blockIdx.x * blockDim.x + threadIdx.x;
  if (e >= E) return;
  const int si = src[e], di = dst[e];
  const float4 l4 = *(const float4*)(el + (size_t)si * NHEADS);
  const float4 r4 = *(const float4*)(er + (size_t)di * NHEADS);
  float4 v;
  v.x = lrelu(l4.x + r4.x);
  v.y = lrelu(l4.y + r4.y);
  v.z = lrelu(l4.z + r4.z);
  v.w = lrelu(l4.w + r4.w);
  *(float4*)(ebuf + (size_t)e * NHEADS) = v;
  float* mp = m + (size_t)di * NHEADS;
  atomicMaxF(mp + 0, v.x);
  atomicMaxF(mp + 1, v.y);
  atomicMaxF(mp + 2, v.z);
  atomicMaxF(mp + 3, v.w);
}

// ---------------------------------------------------------------------------
// pass 2 over edges: ex = exp(e - m[dst]); store ex; segment sum
// ---------------------------------------------------------------------------
__global__ void edge_exp_sum(const float* __restrict__ m,
                             const int* __restrict__ dst,
                             float* __restrict__ ebuf,
                             float* __restrict__ s, int E) {
  const int e = blockIdx.x * blockDim.x + threadIdx.x;
  if (e >= E) return;
  const int di = dst[e];
  float4 ev = *(const float4*)(ebuf + (size_t)e * NHEADS);
  const float4 mv = *(const float4*)(m + (size_t)di * NHEADS);
  ev.x = expf(ev.x - mv.x);
  ev.y = expf(ev.y - mv.y);
  ev.z = expf(ev.z - mv.z);
  ev.w = expf(ev.w - mv.w);
  *(float4*)(ebuf + (size_t)e * NHEADS) = ev;
  float* sp = s + (size_t)di * NHEADS;
  atomAddF(sp + 0, ev.x);
  atomAddF(sp + 1, ev.y);
  atomAddF(sp + 2, ev.z);
  atomAddF(sp + 3, ev.w);
}

// ---------------------------------------------------------------------------
// pass 3 over edges: out[dst] += feat[src] * alpha. One wave per edge,
// 8 consecutive floats per lane (each lane stays inside one head).
// ---------------------------------------------------------------------------
__global__ __launch_bounds__(256) void edge_scatter(
    const float* __restrict__ feat, const float* __restrict__ ebuf,
    const float* __restrict__ s, const int* __restrict__ src,
    const int* __restrict__ dst, float* __restrict__ accum, int E) {
  const int lane = threadIdx.x & 31;
  const int e = blockIdx.x * 8 + (threadIdx.x >> 5);
  if (e >= E) return;
  const int si = src[e], di = dst[e];
  const int base = lane * 8;
  const int hh = base >> 6;  // head for this lane's 8 floats
  const float alpha =
      ebuf[(size_t)e * NHEADS + hh] / s[(size_t)di * NHEADS + hh];
  const float* fp = feat + (size_t)si * HID + base;
  const float4 f0 = *(const float4*)(fp);
  const float4 f1 = *(const float4*)(fp + 4);
  float* op = accum + (size_t)di * HID + base;
  atomAddF(op + 0, f0.x * alpha);
  atomAddF(op + 1, f0.y * alpha);
  atomAddF(op + 2, f0.z * alpha);
  atomAddF(op + 3, f0.w * alpha);
  atomAddF(op + 4, f1.x * alpha);
  atomAddF(op + 5, f1.y * alpha);
  atomAddF(op + 6, f1.z * alpha);
  atomAddF(op + 7, f1.w * alpha);
}

// ---------------------------------------------------------------------------
// residual + relu epilogue (in place OK)
// ---------------------------------------------------------------------------
__global__ void epilogue_relu(const float* __restrict__ accum,
                              const float* __restrict__ hprev,
                              float* __restrict__ hnext, int n, int residual) {
  const int i = blockIdx.x * blockDim.x + threadIdx.x;
  if (i >= n) return;
  float v = accum[i] + (residual ? hprev[i] : 0.0f);
  hnext[i] = v > 0.0f ? v : 0.0f;
}

// ---------------------------------------------------------------------------
// final layer: logits[n,d] = mean_h(accum[n,h,d] + hprev[n,h,d])
// ---------------------------------------------------------------------------
__global__ void final_mean(const float* __restrict__ accum,
                           const float* __restrict__ hprev,
                           float* __restrict__ logits, int ND) {
  const int i = blockIdx.x * blockDim.x + threadIdx.x;
  if (i >= ND) return;
  const int n = i >> 6, d = i & (DHEAD - 1);
  const float* ap = accum + (size_t)n * HID + d;
  const float* hp = hprev + (size_t)n * HID + d;
  float v = 0.0f;
#pragma unroll
  for (int h = 0; h < NHEADS; ++h) v += ap[h * DHEAD] + hp[h * DHEAD];
  logits[i] = v * 0.25f;
}

// ---------------------------------------------------------------------------
// host orchestration
// ---------------------------------------------------------------------------
static void run_layer(const float* h, int K, const float* W, const float* al,
                      const float* ar, const int* src, const int* dst,
                      float* feat, float* ebuf, float* el, float* er, float* m,
                      float* s, float* accum, int N, int E,
                      hipStream_t stream) {
  const int rowTiles = (N + 15) / 16;
  dim3 gg((rowTiles + 7) / 8, HID / 64);
  gemm_wmma_f32<<<gg, 256, 0, stream>>>(h, W, feat, N, K);

  const int NH = N * NHEADS;
  node_attn<<<(NH + 7) / 8, 256, 0, stream>>>(feat, al, ar, el, er, NH);

  const int nOut = N * HID;
  init_bufs<<<(nOut + 255) / 256, 256, 0, stream>>>(accum, m, s, nOut, NH);

  edge_logits_max<<<(E + 255) / 256, 256, 0, stream>>>(el, er, src, dst, ebuf,
                                                       m, E);
  edge_exp_sum<<<(E + 255) / 256, 256, 0, stream>>>(m, dst, ebuf, s, E);
  edge_scatter<<<(E + 7) / 8, 256, 0, stream>>>(feat, ebuf, s, src, dst, accum,
                                                E);
}

extern "C" void kernel_launch(void* const* d_in, const int* in_sizes, int n_in,
                              void* d_out, int out_size, void* d_ws,
                              size_t ws_size, hipStream_t stream) {
  const float* inputs = (const float*)d_in[0];
  const int* src = (const int*)d_in[1];
  const int* dst = (const int*)d_in[2];
  const float* W0 = (const float*)d_in[3];
  const float* al0 = (const float*)d_in[4];
  const float* ar0 = (const float*)d_in[5];
  const float* W1 = (const float*)d_in[6];
  const float* al1 = (const float*)d_in[7];
  const float* ar1 = (const float*)d_in[8];
  const float* W2 = (const float*)d_in[9];
  const float* al2 = (const float*)d_in[10];
  const float* ar2 = (const float*)d_in[11];

  const int IN = 128;
  const int N = in_sizes[0] / IN;   // 50000
  const int E = in_sizes[1];        // 800000
  const size_t nhid = (size_t)N * HID;
  const size_t nh = (size_t)N * NHEADS;

  float* ws = (float*)d_ws;
  float* bufA = ws;                       // feat scratch  [N*HID]
  float* bufB = bufA + nhid;              // accum / h     [N*HID]
  float* bufC = bufB + nhid;              // accum / h     [N*HID]
  float* ebuf = bufC + nhid;              // edge ex       [E*H]
  float* el = ebuf + (size_t)E * NHEADS;  // [N*H]
  float* er = el + nh;
  float* m = er + nh;
  float* s = m + nh;

  float* logits = (float*)d_out;
  const int nOut = N * HID;
  const int ND = N * DHEAD;

  // layer 0: inputs[N,128] -> bufB, relu, no residual
  run_layer(inputs, IN, W0, al0, ar0, src, dst, bufA, ebuf, el, er, m, s, bufB,
            N, E, stream);
  epilogue_relu<<<(nOut + 255) / 256, 256, 0, stream>>>(bufB, bufB, bufB, nOut,
                                                        0);
  // layer 1: bufB -> bufC, relu, identity residual
  run_layer(bufB, HID, W1, al1, ar1, src, dst, bufA, ebuf, el, er, m, s, bufC,
            N, E, stream);
  epilogue_relu<<<(nOut + 255) / 256, 256, 0, stream>>>(bufC, bufB, bufC, nOut,
                                                        1);
  // layer 2: bufC -> bufB, residual, no activation, mean over heads
  run_layer(bufC, HID, W2, al2, ar2, src, dst, bufA, ebuf, el, er, m, s, bufB,
            N, E, stream);
  final_mean<<<(ND + 255) / 256, 256, 0, stream>>>(bufB, bufC, logits, ND);
}